// MyGCN2Defect_27642409517487
// MI455X (gfx1250) — compile-verified
//
#include <hip/hip_runtime.h>

#define FD 18
#define KPAD 20   // K padded 18 -> 20 (five 16x16x4 steps)
#define NPAD 32   // N padded 18 -> 32 (two 16-wide N tiles)

typedef __attribute__((ext_vector_type(2))) float v2f;
typedef __attribute__((ext_vector_type(8))) float v8f;

__global__ void gcn_zero_kernel(float* __restrict__ p, long n) {
    long i = (long)blockIdx.x * blockDim.x + threadIdx.x;
    long stride = (long)gridDim.x * blockDim.x;
    for (; i < n; i += stride) p[i] = 0.0f;
}

__global__ void gcn_degree_kernel(const int* __restrict__ src, const int* __restrict__ dst,
                                  float* __restrict__ degOut, float* __restrict__ degIn, int E) {
    int e = blockIdx.x * blockDim.x + threadIdx.x;
    if (e >= E) return;
    unsafeAtomicAdd(&degOut[src[e]], 1.0f);
    unsafeAtomicAdd(&degIn[dst[e]], 1.0f);
}

// One wave32 per 16-node tile. Full fp32 WMMA (V_WMMA_F32_16X16X4_F32):
//   h   = (feat * rsqrt(max(degOut,1))) @ W_conv      [16 x 18]
//   lin = feat @ W_lin                                 [16 x 18]
// Weights staged in LDS as zero-padded 20x32 tiles -> branch-free inner loop.
__global__ __launch_bounds__(256, 2)
void gcn_gemm_wmma_kernel(const float* __restrict__ feat, const float* __restrict__ degOut,
                          const float* __restrict__ Wc, const float* __restrict__ Wl,
                          float* __restrict__ h, float* __restrict__ lin,
                          int nTiles, int N) {
    __shared__ float sWc[KPAD * NPAD];
    __shared__ float sWl[KPAD * NPAD];
    for (int i = threadIdx.x; i < KPAD * NPAD; i += blockDim.x) {
        const int r = i >> 5;          // 0..19
        const int c = i & 31;          // 0..31
        const bool ok = (r < FD) && (c < FD);
        sWc[i] = ok ? Wc[r * FD + c] : 0.0f;
        sWl[i] = ok ? Wl[r * FD + c] : 0.0f;
    }
    __syncthreads();

    const int wave = (int)((blockIdx.x * (unsigned)blockDim.x + threadIdx.x) >> 5);
    if (wave >= nTiles) return;                 // wave-uniform: EXEC stays all-1
    const int lane  = threadIdx.x & 31;
    const int m     = lane & 15;                // A: matrix row, B: matrix col
    const int khalf = (lane >> 4) << 1;         // 0 for lanes 0-15, 2 for lanes 16-31
    const int node  = wave * 16 + m;
    const bool nodeOk = (node < N);
    const int nodeC = nodeOk ? node : (N - 1);
    const float scale = nodeOk ? rsqrtf(fmaxf(degOut[nodeC], 1.0f)) : 0.0f;
    const float* frow = feat + (long)nodeC * FD;
    const int c1 = 16 + m;                      // column in second N-tile (16..31, LDS zero-padded)

    v8f aC0 = {}; v8f aC1 = {}; v8f aL0 = {}; v8f aL1 = {};
    #pragma unroll
    for (int k0 = 0; k0 < KPAD; k0 += 4) {
        const int k = k0 + khalf;               // even: 0..18
        const bool kok = (k < FD);              // k==18 -> zero pad (k+1<=17 when kok)
        const int kc = kok ? k : 0;

        // A-matrix fragment (16x4): lane holds feat[node][k], feat[node][k+1]
        float fx = frow[kc];
        float fy = frow[kc + 1];
        v2f af; af.x = (kok && nodeOk) ? fx : 0.0f;
                af.y = (kok && nodeOk) ? fy : 0.0f;
        v2f as; as.x = af.x * scale; as.y = af.y * scale;

        // B-matrix fragments (4x16) from zero-padded LDS: rows k, k+1 (k+1 <= 19)
        v2f bC0, bL0, bC1, bL1;
        bC0.x = sWc[k * NPAD + m];        bC0.y = sWc[(k + 1) * NPAD + m];
        bL0.x = sWl[k * NPAD + m];        bL0.y = sWl[(k + 1) * NPAD + m];
        bC1.x = sWc[k * NPAD + c1];       bC1.y = sWc[(k + 1) * NPAD + c1];
        bL1.x = sWl[k * NPAD + c1];       bL1.y = sWl[(k + 1) * NPAD + c1];

        aC0 = __builtin_amdgcn_wmma_f32_16x16x4_f32(false, as, false, bC0, (short)0, aC0, false, false);
        aC1 = __builtin_amdgcn_wmma_f32_16x16x4_f32(false, as, false, bC1, (short)0, aC1, false, false);
        aL0 = __builtin_amdgcn_wmma_f32_16x16x4_f32(false, af, false, bL0, (short)0, aL0, false, false);
        aL1 = __builtin_amdgcn_wmma_f32_16x16x4_f32(false, af, false, bL1, (short)0, aL1, false, false);
    }

    // C/D layout: VGPR v, lanes 0-15 -> M=v, lanes 16-31 -> M=v+8; N = lane&15
    const int nCol  = lane & 15;
    const int mBase = (lane >> 4) * 8;
    #pragma unroll
    for (int v = 0; v < 8; ++v) {
        const int row = wave * 16 + mBase + v;
        if (row < N) {
            float* hr = h   + (long)row * FD;
            float* lr = lin + (long)row * FD;
            hr[nCol] = aC0[v];
            lr[nCol] = aL0[v];
            if (nCol < FD - 16) {               // second N-tile: cols 16..17
                hr[16 + nCol] = aC1[v];
                lr[16 + nCol] = aL1[v];
            }
        }
    }
}

__global__ void gcn_scatter_kernel(const int* __restrict__ src, const int* __restrict__ dst,
                                   const float* __restrict__ h, float* __restrict__ agg, int E) {
    int e = blockIdx.x * blockDim.x + threadIdx.x;
    if (e >= E) return;
    const float* hr = h + (long)src[e] * FD;    // 8-byte aligned (72B rows)
    float* ar = agg + (long)dst[e] * FD;
    #pragma unroll
    for (int j = 0; j < FD; j += 2) {
        v2f v = *(const v2f*)(hr + j);
        unsafeAtomicAdd(ar + j,     v.x);
        unsafeAtomicAdd(ar + j + 1, v.y);
    }
}

__global__ void gcn_final_kernel(const float* __restrict__ agg, const float* __restrict__ degIn,
                                 const float* __restrict__ lin,
                                 const float* __restrict__ bConv, const float* __restrict__ bLin,
                                 const float* __restrict__ Wcls, const float* __restrict__ bCls,
                                 float* __restrict__ out, int N) {
    int n = blockIdx.x * blockDim.x + threadIdx.x;
    if (n >= N) return;
    const float nd = rsqrtf(fmaxf(degIn[n], 1.0f));
    const float* ar = agg + (long)n * FD;
    const float* lr = lin + (long)n * FD;
    float acc0 = bCls[0], acc1 = bCls[1];
    #pragma unroll
    for (int j = 0; j < FD; ++j) {              // relu(lin + b_lin) part of cat
        float lv = fmaxf(lr[j] + bLin[j], 0.0f);
        acc0 += lv * Wcls[j * 2 + 0];
        acc1 += lv * Wcls[j * 2 + 1];
    }
    #pragma unroll
    for (int j = 0; j < FD; ++j) {              // relu(conv_out) part (relu idempotent)
        float cv = fmaxf(ar[j] * nd + bConv[j], 0.0f);
        acc0 += cv * Wcls[(FD + j) * 2 + 0];
        acc1 += cv * Wcls[(FD + j) * 2 + 1];
    }
    out[n * 2 + 0] = acc0;
    out[n * 2 + 1] = acc1;
}

extern "C" void kernel_launch(void* const* d_in, const int* in_sizes, int n_in,
                              void* d_out, int out_size, void* d_ws, size_t ws_size,
                              hipStream_t stream) {
    const float* feat = (const float*)d_in[0];
    const int*   src  = (const int*)d_in[1];
    const int*   dst  = (const int*)d_in[2];
    const float* Wc   = (const float*)d_in[3];
    const float* bC   = (const float*)d_in[4];
    const float* Wl   = (const float*)d_in[5];
    const float* bL   = (const float*)d_in[6];
    const float* Wcl  = (const float*)d_in[7];
    const float* bCl  = (const float*)d_in[8];
    const int N = in_sizes[0] / FD;
    const int E = in_sizes[1];

    float* degOut = (float*)d_ws;
    float* degIn  = degOut + N;
    float* h      = degIn + N;
    float* lin    = h   + (size_t)N * FD;
    float* agg    = lin + (size_t)N * FD;

    gcn_zero_kernel<<<512, 256, 0, stream>>>(degOut, (long)2 * N);        // deg_out + deg_in
    gcn_zero_kernel<<<2048, 256, 0, stream>>>(agg, (long)N * FD);
    gcn_degree_kernel<<<(E + 255) / 256, 256, 0, stream>>>(src, dst, degOut, degIn, E);
    const int nTiles = (N + 15) / 16;
    gcn_gemm_wmma_kernel<<<(nTiles + 7) / 8, 256, 0, stream>>>(feat, degOut, Wc, Wl, h, lin, nTiles, N);
    gcn_scatter_kernel<<<(E + 255) / 256, 256, 0, stream>>>(src, dst, h, agg, E);
    gcn_final_kernel<<<(N + 255) / 256, 256, 0, stream>>>(agg, degIn, lin, bC, bL, Wcl, bCl, (float*)d_out, N);
}